// GlobalTopKGating_26061861552656
// MI455X (gfx1250) — compile-verified
//
#include <hip/hip_runtime.h>
#include <math.h>

// ---------------------------------------------------------------------------
// GlobalTopKGating on MI455X (gfx1250, wave32).
//
// Roofline: avg-pool reads 201 MB (bound: ~8.6us @ 23.3 TB/s HBM). GEMMs are
// ~340 MFLOP on 1.2 MB of L2-resident data -> compute is free; keep fp32 and
// use V_WMMA_F32_16X16X4_F32 so the gating top-k is numerically exact.
// ---------------------------------------------------------------------------

typedef __attribute__((ext_vector_type(2))) float v2f;
typedef __attribute__((ext_vector_type(8))) float v8f;

#define BN_EPS 1e-5f
#define TEMP   2.0f

#define B_   64
#define C_   768
#define C2_  1536
#define R_   96
#define E_   16
#define HW_  1024

__device__ __forceinline__ float gelu_exact(float x) {
  // torch-default exact GELU: 0.5*x*(1+erf(x/sqrt(2)))
  return 0.5f * x * (1.0f + erff(x * 0.7071067811865475f));
}

// ---------------------------------------------------------------------------
// Kernel 1: global average pool [B,C,H,W] -> [B,C]. One wave per (b,c).
// ---------------------------------------------------------------------------
__global__ void avgpool_kernel(const float* __restrict__ x,
                               float* __restrict__ g) {
  const int wave = (blockIdx.x * blockDim.x + threadIdx.x) >> 5;
  const int lane = threadIdx.x & 31;
  if (wave >= B_ * C_) return;

  const float4* p4 = (const float4*)(x + (size_t)wave * HW_);
  // hint the next channel's line into cache (lowers to global_prefetch_b8)
  __builtin_prefetch(x + (size_t)(wave + 1) * HW_, 0, 1);

  float s = 0.f;
#pragma unroll
  for (int i = 0; i < HW_ / 4 / 32; ++i) {      // 8 x float4 per lane
    float4 v = p4[lane + i * 32];
    s += (v.x + v.y) + (v.z + v.w);
  }
#pragma unroll
  for (int off = 16; off > 0; off >>= 1)        // wave32 butterfly reduce
    s += __shfl_xor(s, off, 32);
  if (lane == 0) g[wave] = s * (1.0f / HW_);
}

// ---------------------------------------------------------------------------
// Wave-level fp32 WMMA tile: D(16x16) = A[tm-tile] * W^T[tn-tile], K step 4.
//   A: [M,K] row-major.  W: [N,K] row-major (torch Linear weight -> x @ W^T).
// fp32 A frag (16x4): lane half h holds K pair {2h, 2h+1} for row (lane&15).
// fp32 B frag (4x16): lane half h holds K pair {2h, 2h+1} for col (lane&15),
// which is a contiguous float2 of W's row -> both fragments are v2f loads.
// EXEC must be all 1s: callers guard at wave granularity only.
// ---------------------------------------------------------------------------
__device__ __forceinline__ v8f wmma_tile_f32(const float* __restrict__ A,
                                             const float* __restrict__ W,
                                             int K, int tm, int tn) {
  const int lane = threadIdx.x & 31;
  const int half = lane >> 4;
  const int l    = lane & 15;
  const float* arow = A + (size_t)(tm * 16 + l) * K + 2 * half;
  const float* wrow = W + (size_t)(tn * 16 + l) * K + 2 * half;
  v8f acc = {0.f, 0.f, 0.f, 0.f, 0.f, 0.f, 0.f, 0.f};
  for (int k = 0; k < K; k += 4) {
    v2f a = *(const v2f*)(arow + k);
    v2f b = *(const v2f*)(wrow + k);
    acc = __builtin_amdgcn_wmma_f32_16x16x4_f32(
        /*neg_a=*/false, a, /*neg_b=*/false, b,
        /*c_mod=*/(short)0, acc, /*reuse_a=*/false, /*reuse_b=*/false);
  }
  return acc;
}

// D layout (fp32 16x16 C/D): element r of acc -> row (tm*16 + r + 8*half),
// col (tn*16 + (lane&15)).

// ---------------------------------------------------------------------------
// Kernel 2: out = gelu(BN(A @ W^T + bias))           (used for w1 and w2)
// ---------------------------------------------------------------------------
__global__ void gemm_bn_gelu_kernel(const float* __restrict__ A,
                                    const float* __restrict__ W,
                                    const float* __restrict__ bias,
                                    const float* __restrict__ bn_g,
                                    const float* __restrict__ bn_b,
                                    const float* __restrict__ bn_m,
                                    const float* __restrict__ bn_v,
                                    float* __restrict__ out,
                                    int N, int K) {
  const int wave   = blockIdx.x * (blockDim.x >> 5) + (threadIdx.x >> 5);
  const int tilesN = N >> 4;
  if (wave >= (B_ >> 4) * tilesN) return;       // wave-uniform guard
  const int tm = wave / tilesN, tn = wave % tilesN;

  v8f acc = wmma_tile_f32(A, W, K, tm, tn);

  const int lane = threadIdx.x & 31, half = lane >> 4, l = lane & 15;
  const int col = tn * 16 + l;
  const float sc = rsqrtf(bn_v[col] + BN_EPS) * bn_g[col];
  const float sh = bn_b[col] - bn_m[col] * sc;
  const float bs = bias[col];
#pragma unroll
  for (int r = 0; r < 8; ++r) {
    const int row = tm * 16 + r + 8 * half;
    out[(size_t)row * N + col] = gelu_exact((acc[r] + bs) * sc + sh);
  }
}

// ---------------------------------------------------------------------------
// Kernel 3: out = gelu(A @ W^T + bias)               (ChannelAttention fc1)
// ---------------------------------------------------------------------------
__global__ void gemm_bias_gelu_kernel(const float* __restrict__ A,
                                      const float* __restrict__ W,
                                      const float* __restrict__ bias,
                                      float* __restrict__ out,
                                      int N, int K) {
  const int wave   = blockIdx.x * (blockDim.x >> 5) + (threadIdx.x >> 5);
  const int tilesN = N >> 4;
  if (wave >= (B_ >> 4) * tilesN) return;
  const int tm = wave / tilesN, tn = wave % tilesN;

  v8f acc = wmma_tile_f32(A, W, K, tm, tn);

  const int lane = threadIdx.x & 31, half = lane >> 4, l = lane & 15;
  const int col = tn * 16 + l;
  const float bs = bias[col];
#pragma unroll
  for (int r = 0; r < 8; ++r) {
    const int row = tm * 16 + r + 8 * half;
    out[(size_t)row * N + col] = gelu_exact(acc[r] + bs);
  }
}

// ---------------------------------------------------------------------------
// Kernel 4: out = h * sigmoid(2*(A @ W^T + bias))    (ChannelAttention fc2;
//           avg_out == max_out on a 1x1 map, hence the factor 2)
// ---------------------------------------------------------------------------
__global__ void gemm_gate_kernel(const float* __restrict__ A,
                                 const float* __restrict__ W,
                                 const float* __restrict__ bias,
                                 const float* __restrict__ h,
                                 float* __restrict__ out,
                                 int N, int K) {
  const int wave   = blockIdx.x * (blockDim.x >> 5) + (threadIdx.x >> 5);
  const int tilesN = N >> 4;
  if (wave >= (B_ >> 4) * tilesN) return;
  const int tm = wave / tilesN, tn = wave % tilesN;

  v8f acc = wmma_tile_f32(A, W, K, tm, tn);

  const int lane = threadIdx.x & 31, half = lane >> 4, l = lane & 15;
  const int col = tn * 16 + l;
  const float bs = bias[col];
#pragma unroll
  for (int r = 0; r < 8; ++r) {
    const int row = tm * 16 + r + 8 * half;
    const float a2 = 2.0f * (acc[r] + bs);
    const float sg = 1.0f / (1.0f + expf(-a2));
    out[(size_t)row * N + col] = h[(size_t)row * N + col] * sg;
  }
}

// ---------------------------------------------------------------------------
// Kernel 5: scores = A @ W^T + bias                  (C -> E gating scores)
// ---------------------------------------------------------------------------
__global__ void gemm_bias_kernel(const float* __restrict__ A,
                                 const float* __restrict__ W,
                                 const float* __restrict__ bias,
                                 float* __restrict__ out,
                                 int N, int K) {
  const int wave   = blockIdx.x * (blockDim.x >> 5) + (threadIdx.x >> 5);
  const int tilesN = N >> 4;
  if (wave >= (B_ >> 4) * tilesN) return;
  const int tm = wave / tilesN, tn = wave % tilesN;

  v8f acc = wmma_tile_f32(A, W, K, tm, tn);

  const int lane = threadIdx.x & 31, half = lane >> 4, l = lane & 15;
  const int col = tn * 16 + l;
  const float bs = bias[col];
#pragma unroll
  for (int r = 0; r < 8; ++r) {
    const int row = tm * 16 + r + 8 * half;
    out[(size_t)row * N + col] = acc[r] + bs;
  }
}

// ---------------------------------------------------------------------------
// Kernel 6: per-row top-2 + temperature softmax. Strict '>' keeps the lowest
// index on ties (matches jax.lax.top_k ordering). Output: idx[B,2] as float,
// then probs[B,2], concatenated flat.
// ---------------------------------------------------------------------------
__global__ void topk_softmax_kernel(const float* __restrict__ scores,
                                    float* __restrict__ out) {
  const int b = blockIdx.x * blockDim.x + threadIdx.x;
  if (b >= B_) return;
  const float* s = scores + b * E_;
  float v0 = -INFINITY, v1 = -INFINITY;
  int i0 = 0, i1 = 0;
#pragma unroll
  for (int e = 0; e < E_; ++e) {
    const float v = s[e];
    if (v > v0)      { v1 = v0; i1 = i0; v0 = v; i0 = e; }
    else if (v > v1) { v1 = v; i1 = e; }
  }
  // softmax([v0,v1]/TEMP): p0 = 1/(1+exp((v1-v0)/TEMP))
  const float p0 = 1.0f / (1.0f + expf((v1 - v0) * (1.0f / TEMP)));
  out[2 * b + 0] = (float)i0;
  out[2 * b + 1] = (float)i1;
  out[2 * B_ + 2 * b + 0] = p0;
  out[2 * B_ + 2 * b + 1] = 1.0f - p0;
}

// ---------------------------------------------------------------------------
extern "C" void kernel_launch(void* const* d_in, const int* in_sizes, int n_in,
                              void* d_out, int out_size, void* d_ws, size_t ws_size,
                              hipStream_t stream) {
  const float* x     = (const float*)d_in[0];
  const float* w1    = (const float*)d_in[1];
  const float* b1    = (const float*)d_in[2];
  const float* bn1_g = (const float*)d_in[3];
  const float* bn1_b = (const float*)d_in[4];
  const float* bn1_m = (const float*)d_in[5];
  const float* bn1_v = (const float*)d_in[6];
  const float* caw1  = (const float*)d_in[7];
  const float* cab1  = (const float*)d_in[8];
  const float* caw2  = (const float*)d_in[9];
  const float* cab2  = (const float*)d_in[10];
  const float* w2    = (const float*)d_in[11];
  const float* b2    = (const float*)d_in[12];
  const float* bn2_g = (const float*)d_in[13];
  const float* bn2_b = (const float*)d_in[14];
  const float* bn2_m = (const float*)d_in[15];
  const float* bn2_v = (const float*)d_in[16];
  const float* w3    = (const float*)d_in[17];
  const float* b3    = (const float*)d_in[18];
  float* out = (float*)d_out;

  // workspace carve (fp32): g | h1 | a1 | hg | h2 | scores  (~1.2 MB total)
  float* g  = (float*)d_ws;
  float* h1 = g  + B_ * C_;
  float* a1 = h1 + B_ * C2_;
  float* hg = a1 + B_ * R_;
  float* h2 = hg + B_ * C2_;
  float* sc = h2 + B_ * C_;

  // 1) avg pool: 49152 waves, 8 waves/block
  avgpool_kernel<<<(B_ * C_) / 8, 256, 0, stream>>>(x, g);

  // 2) h1 = gelu(BN1(g @ w1^T + b1))      tiles: 4 x 96 = 384
  gemm_bn_gelu_kernel<<<384 / 4, 128, 0, stream>>>(
      g, w1, b1, bn1_g, bn1_b, bn1_m, bn1_v, h1, C2_, C_);

  // 3) a1 = gelu(h1 @ caw1^T + cab1)      tiles: 4 x 6 = 24
  gemm_bias_gelu_kernel<<<24 / 4, 128, 0, stream>>>(h1, caw1, cab1, a1, R_, C2_);

  // 4) hg = h1 * sigmoid(2*(a1 @ caw2^T + cab2))   tiles: 384
  gemm_gate_kernel<<<384 / 4, 128, 0, stream>>>(a1, caw2, cab2, h1, hg, C2_, R_);

  // 5) h2 = gelu(BN2(hg @ w2^T + b2))     tiles: 4 x 48 = 192
  gemm_bn_gelu_kernel<<<192 / 4, 128, 0, stream>>>(
      hg, w2, b2, bn2_g, bn2_b, bn2_m, bn2_v, h2, C_, C2_);

  // 6) scores = h2 @ w3^T + b3            tiles: 4 x 1 = 4
  gemm_bias_kernel<<<1, 128, 0, stream>>>(h2, w3, b3, sc, E_, C_);

  // 7) top-2 + softmax(vals/2) -> d_out = [idx(64x2) | probs(64x2)]
  topk_softmax_kernel<<<1, 64, 0, stream>>>(sc, out);
}